// InvResMLP_31061203485293
// MI455X (gfx1250) — compile-verified
//
#include <hip/hip_runtime.h>

typedef float v2f __attribute__((ext_vector_type(2)));
typedef float v4f __attribute__((ext_vector_type(4)));
typedef float v8f __attribute__((ext_vector_type(8)));
typedef int   v2i __attribute__((ext_vector_type(2)));

#define BN_EPS 1e-5f

constexpr int B_ = 8, C_ = 64, N_ = 16384, K_ = 16, M_ = 256;

static __device__ __forceinline__ v8f wmma4(v2f a, v2f b, v8f c) {
  // D = A(16x4 f32) * B(4x16 f32) + C(16x16 f32)
  return __builtin_amdgcn_wmma_f32_16x16x4_f32(false, a, false, b, (short)0, c,
                                               false, false);
}

// ---------------------------------------------------------------------------
// k1: x1t[b][n][c] = relu(bn1(W1 @ f))   (output transposed for fast gather)
// grid = (N/32, B), block = 256 (8 waves); wave tile = 16 out-ch x 16 points
// x1t stores use default (RT) policy: we WANT them resident in L2 for k2.
// ---------------------------------------------------------------------------
__global__ __launch_bounds__(256) void k1_pwconv(
    const float* __restrict__ f, const float* __restrict__ W1,
    const float* __restrict__ g1, const float* __restrict__ b1,
    const float* __restrict__ m1, const float* __restrict__ v1,
    float* __restrict__ x1t) {
  __shared__ float s_inv[C_], s_beta[C_];
  const int tid = threadIdx.x;
  if (tid < C_) {
    float inv = g1[tid] * rsqrtf(v1[tid] + BN_EPS);
    s_inv[tid] = inv;
    s_beta[tid] = b1[tid] - m1[tid] * inv;
  }
  __syncthreads();

  const int b = blockIdx.y;
  const int n0 = blockIdx.x * 32;
  const int wave = tid >> 5, lane = tid & 31;
  const int lane15 = lane & 15, hi = lane >> 4;
  const int mbase = (wave & 3) * 16;                 // output-channel block
  const int col = n0 + (wave >> 2) * 16 + lane15;    // point column

  const float* Wrow = W1 + (mbase + lane15) * C_;    // A: row = lane15
  const float* fcol = f + (size_t)b * C_ * N_ + col; // B: col = lane15

  v8f acc = {};
#pragma unroll
  for (int kb = 0; kb < C_; kb += 4) {
    const int k0 = kb + 2 * hi;                      // K = vgpr + 2*hi
    v2f a, bm;
    a.x = Wrow[k0];
    a.y = Wrow[k0 + 1];
    bm.x = fcol[(size_t)k0 * N_];
    bm.y = fcol[(size_t)(k0 + 1) * N_];
    acc = wmma4(a, bm, acc);
  }

  // BN + ReLU, store transposed; rows cbase..cbase+7 are contiguous channels
  const int cbase = mbase + 8 * hi;
  float r[8];
#pragma unroll
  for (int i = 0; i < 8; ++i) {
    const int ch = cbase + i;
    r[i] = fmaxf(acc[i] * s_inv[ch] + s_beta[ch], 0.f);
  }
  float* dst = x1t + ((size_t)b * N_ + col) * C_ + cbase;
  *(v4f*)dst = (v4f){r[0], r[1], r[2], r[3]};
  *(v4f*)(dst + 4) = (v4f){r[4], r[5], r[6], r[7]};
}

// ---------------------------------------------------------------------------
// k2: fused  gather+maxpool(pe + x1t[idx])  ->  relu(bn(Wp1@y))
//            ->  bn(Wp2@h) + f  ->  relu   ->  out[B,C,N]
// grid = (N/32, B), block = 256 (8 waves), 32 points per workgroup
// One-shot streams (pe, idx, out) use non-temporal policy so the 32 MB x1t
// gather working set stays resident in the 192 MB L2.
// ---------------------------------------------------------------------------
__global__ __launch_bounds__(256) void k2_fused(
    const float* __restrict__ pe, const int* __restrict__ idx,
    const float* __restrict__ x1t, const float* __restrict__ f,
    const float* __restrict__ Wp1, const float* __restrict__ gp1,
    const float* __restrict__ bp1, const float* __restrict__ mp1,
    const float* __restrict__ vp1, const float* __restrict__ Wp2,
    const float* __restrict__ gp2, const float* __restrict__ bp2,
    const float* __restrict__ mp2, const float* __restrict__ vp2,
    float* __restrict__ out) {
  __shared__ float yb[C_][33];   // y (k-major), pad 33: conflict-free writes
  __shared__ float hb[M_][40];   // h (k-major), pad 40: conflict-free B reads
  __shared__ int sidx[32 * K_];
  __shared__ float s_inv1[M_], s_beta1[M_], s_inv2[C_], s_beta2[C_];

  const int tid = threadIdx.x;
  const int b = blockIdx.y;
  const int n0 = blockIdx.x * 32;

  {  // fold BN params: z*inv + beta
    float inv = gp1[tid] * rsqrtf(vp1[tid] + BN_EPS);
    s_inv1[tid] = inv;
    s_beta1[tid] = bp1[tid] - mp1[tid] * inv;
  }
  if (tid < C_) {
    float inv = gp2[tid] * rsqrtf(vp2[tid] + BN_EPS);
    s_inv2[tid] = inv;
    s_beta2[tid] = bp2[tid] - mp2[tid] * inv;
  }
  {  // stage neighbor indices (32 pts x 16 nbrs), streamed non-temporally
    const v2i* ib = (const v2i*)(idx + ((size_t)b * N_ + n0) * K_);
    v2i two = __builtin_nontemporal_load(ib + tid);
    sidx[2 * tid] = two.x;
    sidx[2 * tid + 1] = two.y;
  }
  __syncthreads();

  // Phase A: y[c][n] = max_k( pe[b,c,n,k] + x1t[b, idx[n,k], c] )
  {
    const int c = tid & 63;
    const float* x1b = x1t + (size_t)b * N_ * C_ + c;
    for (int n = tid >> 6; n < 32; n += 4) {
      const v4f* pep =
          (const v4f*)(pe + (((size_t)b * C_ + c) * N_ + (n0 + n)) * K_);
      const int* in = sidx + n * K_;
      float acc = -3.4e38f;
#pragma unroll
      for (int k4 = 0; k4 < 4; ++k4) {
        v4f pv = __builtin_nontemporal_load(pep + k4);  // stream pe, keep L2
#pragma unroll                                          // for x1t gathers
        for (int j = 0; j < 4; ++j) {
          float fj = x1b[(size_t)in[k4 * 4 + j] * C_];
          acc = fmaxf(acc, pv[j] + fj);
        }
      }
      yb[c][n] = acc;
    }
  }
  __syncthreads();

  const int wave = tid >> 5, lane = tid & 31;
  const int lane15 = lane & 15, hi = lane >> 4;

  // Phase B: h = relu(bn(Wp1 @ y)),  Wp1[256x64] * y[64x32] -> hb[256x32]
#pragma unroll
  for (int t = 0; t < 4; ++t) {
    const int tile = wave * 4 + t;       // 32 tiles of 16x16
    const int mbase = (tile >> 1) * 16;
    const int colL = (tile & 1) * 16 + lane15;
    const float* Wrow = Wp1 + (mbase + lane15) * C_;
    v8f acc = {};
#pragma unroll
    for (int kb = 0; kb < C_; kb += 4) {
      const int k0 = kb + 2 * hi;
      v2f a, bm;
      a.x = Wrow[k0];
      a.y = Wrow[k0 + 1];
      bm.x = yb[k0][colL];
      bm.y = yb[k0 + 1][colL];
      acc = wmma4(a, bm, acc);
    }
    const int cbase = mbase + 8 * hi;
#pragma unroll
    for (int i = 0; i < 8; ++i) {
      const int ch = cbase + i;
      hb[ch][colL] = fmaxf(acc[i] * s_inv1[ch] + s_beta1[ch], 0.f);
    }
  }
  __syncthreads();

  // Phase C: out = relu(bn(Wp2 @ h) + f),  Wp2[64x256] * h[256x32]
  {
    const int mbase = (wave & 3) * 16;
    const int colL = (wave >> 2) * 16 + lane15;
    const int n = n0 + colL;
    const float* Wrow = Wp2 + (mbase + lane15) * M_;
    v8f acc = {};
#pragma unroll
    for (int kb = 0; kb < M_; kb += 4) {
      const int k0 = kb + 2 * hi;
      v2f a, bm;
      a.x = Wrow[k0];
      a.y = Wrow[k0 + 1];
      bm.x = hb[k0][colL];
      bm.y = hb[k0 + 1][colL];
      acc = wmma4(a, bm, acc);
    }
    const int cbase = mbase + 8 * hi;
#pragma unroll
    for (int i = 0; i < 8; ++i) {
      const int ch = cbase + i;
      const size_t o = ((size_t)b * C_ + ch) * N_ + n;
      float val = acc[i] * s_inv2[ch] + s_beta2[ch] + f[o];  // residual
      __builtin_nontemporal_store(fmaxf(val, 0.f), out + o); // write-once
    }
  }
}

// ---------------------------------------------------------------------------
extern "C" void kernel_launch(void* const* d_in, const int* in_sizes, int n_in,
                              void* d_out, int out_size, void* d_ws,
                              size_t ws_size, hipStream_t stream) {
  (void)in_sizes; (void)n_in; (void)out_size; (void)ws_size;
  // setup_inputs order: p(unused), f, pe, idx, W1, g1,b1,m1,v1,
  //                     Wp1, gp1,bp1,mp1,vp1, Wp2, gp2,bp2,mp2,vp2
  const float* f   = (const float*)d_in[1];
  const float* pe  = (const float*)d_in[2];
  const int*   idx = (const int*)d_in[3];
  const float* W1  = (const float*)d_in[4];
  const float* g1  = (const float*)d_in[5];
  const float* b1  = (const float*)d_in[6];
  const float* m1  = (const float*)d_in[7];
  const float* v1  = (const float*)d_in[8];
  const float* Wp1 = (const float*)d_in[9];
  const float* gp1 = (const float*)d_in[10];
  const float* bp1 = (const float*)d_in[11];
  const float* mp1 = (const float*)d_in[12];
  const float* vp1 = (const float*)d_in[13];
  const float* Wp2 = (const float*)d_in[14];
  const float* gp2 = (const float*)d_in[15];
  const float* bp2 = (const float*)d_in[16];
  const float* mp2 = (const float*)d_in[17];
  const float* vp2 = (const float*)d_in[18];
  float* out = (float*)d_out;
  float* x1t = (float*)d_ws;  // B*N*C fp32 = 32 MiB scratch, [B,N,C]

  dim3 grid(N_ / 32, B_);
  k1_pwconv<<<grid, 256, 0, stream>>>(f, W1, g1, b1, m1, v1, x1t);
  k2_fused<<<grid, 256, 0, stream>>>(pe, idx, x1t, f, Wp1, gp1, bp1, mp1, vp1,
                                     Wp2, gp2, bp2, mp2, vp2, out);
}